// FAST_SAB_90907277787370
// MI455X (gfx1250) — compile-verified
//
#include <hip/hip_runtime.h>
#include <math.h>

// Problem constants (match reference)
#define Bsz 64
#define Tsz 512
#define Isz 128
#define Hsz 512
#define Csz 128
#define Msz 20
#define RMAX 20

typedef __attribute__((ext_vector_type(2))) float v2f;
typedef __attribute__((ext_vector_type(8))) float v8f;

// CDNA5 f32 tensor op: D(16x16) = A(16x4) * B(4x16) + C, full fp32.
__device__ __forceinline__ v8f wmma4(v2f a, v2f b, v8f c) {
  return __builtin_amdgcn_wmma_f32_16x16x4_f32(
      /*neg_a=*/false, a, /*neg_b=*/false, b,
      /*c_mod=*/(short)0, c, /*reuse_a=*/false, /*reuse_b=*/false);
}

// Branch-free K-loop: per step one 8B A-fragment load (contiguous row data) and
// one 8B B-fragment load (pre-packed fragment order), then one WMMA.
//   aptr: lane's A row base + half*2   (advances 4 floats per K-step)
//   bptr: packed B tile base + lane*2  (advances 64 floats per K-step)
__device__ __forceinline__ v8f gemm_k(const float* __restrict__ aptr,
                                      const float* __restrict__ bptr,
                                      int kSteps, v8f acc) {
#pragma unroll 4
  for (int kk = 0; kk < kSteps; ++kk) {
    v2f a = *(const v2f*)(aptr + kk * 4);
    v2f b = *(const v2f*)(bptr + kk * 64);
    acc = wmma4(a, b, acc);
  }
  return acc;
}

__device__ __forceinline__ float sigf(float x) { return 1.0f / (1.0f + expf(-x)); }

// ---------------------------------------------------------------- init
__global__ void k_init(float* __restrict__ p, int n) {
  int i = blockIdx.x * blockDim.x + threadIdx.x;
  if (i < n) p[i] = 0.0f;
}

// ---------------------------------------------------------------- weight pre-pack
// dst flat index i = ((nt*KS + kk)*32 + lane)*2 + j  maps to  B[k][n] with
//   k = kk*4 + (lane>>4)*2 + j,  n = nt*16 + (lane&15)
// MODE 0: B[k][n] = k<Isz ? w_ih[n][k] : w_hh[n][k-Isz]          (s0=w_ih, s1=w_hh)
// MODE 1: B[k][n] = s0[k*Hsz + n]                                 (sel_w1 / w_t half)
// MODE 2: B[k][n] = s0[n*(2*Hsz) + k]                             (fc_w)
template <int MODE>
__global__ void k_pack(const float* __restrict__ s0, const float* __restrict__ s1,
                       float* __restrict__ dst, int K, int N) {
  int i = blockIdx.x * blockDim.x + threadIdx.x;
  if (i >= K * N) return;
  int j = i & 1;
  int lane = (i >> 1) & 31;
  int kk = (i >> 6) % (K / 4);
  int nt = (i >> 6) / (K / 4);
  int k = kk * 4 + (lane >> 4) * 2 + j;
  int n = nt * 16 + (lane & 15);
  float v;
  if (MODE == 0)      v = (k < Isz) ? s0[n * Isz + k] : s1[n * Hsz + (k - Isz)];
  else if (MODE == 1) v = s0[k * Hsz + n];
  else                v = s0[n * (2 * Hsz) + k];
  dst[i] = v;
}

// ---------------------------------------------------------------- gates + pred GEMM
// A = [x_t | h] (64 x 640); nt<128: gates (K=640, packed [w_ih|w_hh]);
// nt>=128: pred (K=512 of sel_w1 only — x-block contributes zero).
__global__ __launch_bounds__(256) void k_gates(
    const float* __restrict__ x, const float* __restrict__ h,
    const float* __restrict__ pWg, const float* __restrict__ pWp,
    const float* __restrict__ b_ih, const float* __restrict__ b_hh,
    float* __restrict__ g, float* __restrict__ predpre, int t) {
  const int NT = 160;
  int wave = (blockIdx.x * blockDim.x + threadIdx.x) >> 5;
  int lane = threadIdx.x & 31;
  int mt = wave / NT, nt = wave % NT;
  if (mt >= Bsz / 16) return;
  int half = lane >> 4, l16 = lane & 15;
  int m = mt * 16 + l16;
  const float* hrow = h + m * Hsz + half * 2;
  v8f acc = {};
  if (nt < 128) {
    const float* xrow = x + (size_t)(m * Tsz + t) * Isz + half * 2;
    const float* bp = pWg + (size_t)nt * (160 * 64) + lane * 2;
    acc = gemm_k(xrow, bp, Isz / 4, acc);                    // K = 0..127   (x @ w_ih.T)
    acc = gemm_k(hrow, bp + (Isz / 4) * 64, Hsz / 4, acc);   // K = 128..639 (h @ w_hh.T)
  } else {
    const float* bp = pWp + (size_t)(nt - 128) * (128 * 64) + lane * 2;
    acc = gemm_k(hrow, bp, Hsz / 4, acc);                    // pred = h @ sel_w1
  }
  int col = nt * 16 + l16;
#pragma unroll
  for (int v = 0; v < 8; v++) {
    int row = mt * 16 + half * 8 + v;
    if (nt < 128) g[row * (4 * Hsz) + col] = acc[v] + b_ih[col] + b_hh[col];
    else          predpre[row * Hsz + (col - 4 * Hsz)] = acc[v];
  }
}

// ---------------------------------------------------------------- LSTM pointwise
__global__ __launch_bounds__(256) void k_lstm(const float* __restrict__ g,
                                              float* __restrict__ c,
                                              float* __restrict__ hl) {
  int i = blockIdx.x * blockDim.x + threadIdx.x;
  if (i >= Bsz * Hsz) return;
  int b = i / Hsz, j = i % Hsz;
  const float* gb = g + b * 4 * Hsz;
  float ig = gb[j], fg = gb[Hsz + j], gg = gb[2 * Hsz + j], og = gb[3 * Hsz + j];
  float cn = sigf(fg) * c[i] + sigf(ig) * tanhf(gg);
  c[i] = cn;
  hl[i] = sigf(og) * tanhf(cn);
}

// ---------------------------------------------------------------- attention projection GEMM
// rows 0..63: u = h_lstm @ w_t[:H]; rows 64..64+64R-1: v = h_old @ w_t[H:]
__global__ __launch_bounds__(256) void k_uv(const float* __restrict__ hl,
                                            const float* __restrict__ h_old,
                                            const float* __restrict__ pWt1,
                                            const float* __restrict__ pWt2,
                                            float* __restrict__ uv, int R) {
  const int NT = Hsz / 16;  // 32
  int wave = (blockIdx.x * blockDim.x + threadIdx.x) >> 5;
  int lane = threadIdx.x & 31;
  int mt = wave / NT, nt = wave % NT;
  if (mt >= 4 + 4 * R) return;
  int half = lane >> 4, l16 = lane & 15;
  const float* Ap;
  if (mt < 4) {
    Ap = hl + (mt * 16 + l16) * Hsz;
  } else {
    int gr = (mt - 4) * 16 + l16;
    int bb = gr / R, rr = gr % R;
    Ap = h_old + (bb * RMAX + rr) * Hsz;
  }
  const float* bp = ((mt < 4) ? pWt1 : pWt2) + (size_t)nt * (128 * 64) + lane * 2;
  v8f acc = {};
  acc = gemm_k(Ap + half * 2, bp, Hsz / 4, acc);
  int col = nt * 16 + l16;
#pragma unroll
  for (int v = 0; v < 8; v++) {
    int row = mt * 16 + half * 8 + v;
    uv[row * Hsz + col] = acc[v];
  }
}

// ---------------------------------------------------------------- a_raw[b,r] = tanh( sum_h tanh(u+v) * w_t2[h] )
__global__ __launch_bounds__(256) void k_attnraw(const float* __restrict__ uv,
                                                 const float* __restrict__ w_t2,
                                                 float* __restrict__ araw, int R) {
  int wave = (blockIdx.x * blockDim.x + threadIdx.x) >> 5;
  int lane = threadIdx.x & 31;
  if (wave >= Bsz * R) return;
  int b = wave / R, r = wave % R;
  const float* u = uv + b * Hsz;
  const float* v = uv + (Bsz + b * R + r) * Hsz;
  float s = 0.0f;
  for (int j = lane; j < Hsz; j += 32) s += tanhf(u[j] + v[j]) * w_t2[j];
  for (int off = 16; off > 0; off >>= 1) s += __shfl_xor(s, off, 32);
  if (lane == 0) araw[b * RMAX + r] = tanhf(s);
}

// ---------------------------------------------------------------- sparse top-k + combine
__global__ __launch_bounds__(256) void k_attend(const float* __restrict__ araw,
                                                const float* __restrict__ h_old,
                                                const float* __restrict__ hl,
                                                float* __restrict__ h,
                                                float* __restrict__ attn_c,
                                                float* __restrict__ a_g, int R) {
  int b = blockIdx.x;
  __shared__ float as[RMAX];
  if (threadIdx.x == 0) {
    float a[RMAX], tmp[RMAX];
    for (int r = 0; r < R; r++) a[r] = araw[b * RMAX + r];
    if (R > 5) {
      for (int r = 0; r < R; r++) tmp[r] = a[r];
      for (int i = 0; i < 6; i++) {  // partial selection sort for (TOP_K+1)-th largest
        int mx = i;
        for (int j = i + 1; j < R; j++) if (tmp[j] > tmp[mx]) mx = j;
        float t0 = tmp[i]; tmp[i] = tmp[mx]; tmp[mx] = t0;
      }
      float delta = tmp[5] - 1e-7f;
      float sum = 0.0f;
      for (int r = 0; r < R; r++) { float w = fmaxf(a[r] - delta, 0.0f); a[r] = w; sum += w; }
      float inv = 1.0f / (sum + 1e-7f);
      for (int r = 0; r < R; r++) a[r] *= inv;
    }
    for (int r = 0; r < R; r++) { as[r] = a[r]; a_g[b * RMAX + r] = a[r]; }
  }
  __syncthreads();
  for (int j = threadIdx.x; j < Hsz; j += blockDim.x) {
    float s = 0.0f;
    for (int r = 0; r < R; r++) s += as[r] * h_old[(b * RMAX + r) * Hsz + j];
    h[b * Hsz + j] = hl[b * Hsz + j] + s;
    attn_c[b * Hsz + j] = s;
  }
}

// ---------------------------------------------------------------- aviz (last 10 steps)
__global__ void k_aviz(const float* __restrict__ a_g, float* __restrict__ dst) {
  int r = threadIdx.x;
  if (r >= RMAX) return;
  float s = 0.0f;
  for (int b = 0; b < Bsz; b++) s += a_g[b * RMAX + r];
  dst[r] = s * (1.0f / (float)Bsz);
}

// ---------------------------------------------------------------- state selection
__global__ __launch_bounds__(256) void k_select(const float* __restrict__ predpre,
                                                const float* __restrict__ h,
                                                float* __restrict__ h_old,
                                                float* __restrict__ norms,
                                                float* __restrict__ l2arr, int t) {
  int b = blockIdx.x, tid = threadIdx.x;
  __shared__ float red[256];
  float s = 0.0f;
  for (int j = tid; j < Hsz; j += 256) {
    float d = (t == 0) ? h[b * Hsz + j] : (tanhf(predpre[b * Hsz + j]) - h[b * Hsz + j]);
    s += d * d;
  }
  red[tid] = s; __syncthreads();
  for (int off = 128; off > 0; off >>= 1) { if (tid < off) red[tid] += red[tid + off]; __syncthreads(); }
  __shared__ float l2sh;
  if (tid == 0) { l2sh = sqrtf(red[0]); l2arr[t * Bsz + b] = l2sh; }
  __syncthreads();
  float l2 = l2sh;
  float* hob = h_old + b * RMAX * Hsz;
  float* nb = norms + b * RMAX;
  if (t == 0) {
    if (tid == 0) nb[0] = l2;
    for (int j = tid; j < Hsz; j += 256) hob[j] = h[b * Hsz + j];
    return;
  }
  __shared__ int sp, smode;
  if (tid == 0) {
    if (t >= Msz) {  // full: replace min if l2 > norms[0], then re-insert sorted
      if (l2 > nb[0]) {
        int p = 0;
        while (p + 1 < RMAX && nb[p + 1] < l2) p++;
        sp = p; smode = 1;
      } else smode = 0;
    } else {         // append into sorted list of size t
      int R = t, p = 0;
      while (p < R && nb[p] <= l2) p++;
      sp = p; smode = 2;
    }
  }
  __syncthreads();
  int mode = smode, p = sp;
  if (mode == 0) return;
  if (mode == 1) {
    for (int j = 1; j <= p; j++) {
      for (int k = tid; k < Hsz; k += 256) hob[(j - 1) * Hsz + k] = hob[j * Hsz + k];
      if (tid == 0) nb[j - 1] = nb[j];
      __syncthreads();
    }
  } else {
    int R = t;
    for (int j = R - 1; j >= p; j--) {
      for (int k = tid; k < Hsz; k += 256) hob[(j + 1) * Hsz + k] = hob[j * Hsz + k];
      if (tid == 0) nb[j + 1] = nb[j];
      __syncthreads();
    }
  }
  for (int k = tid; k < Hsz; k += 256) hob[p * Hsz + k] = h[b * Hsz + k];
  if (tid == 0) nb[p] = l2;
}

// ---------------------------------------------------------------- per-step FC GEMM → d_out
__global__ __launch_bounds__(256) void k_fc(const float* __restrict__ h,
                                            const float* __restrict__ attn_c,
                                            const float* __restrict__ pWfc,
                                            const float* __restrict__ fc_b,
                                            float* __restrict__ out, int t) {
  const int NT = Csz / 16;  // 8
  int wave = (blockIdx.x * blockDim.x + threadIdx.x) >> 5;
  int lane = threadIdx.x & 31;
  int mt = wave / NT, nt = wave % NT;
  if (mt >= Bsz / 16) return;
  int half = lane >> 4, l16 = lane & 15;
  int m = mt * 16 + l16;
  const float* bp = pWfc + (size_t)nt * (256 * 64) + lane * 2;
  v8f acc = {};
  acc = gemm_k(h + m * Hsz + half * 2, bp, Hsz / 4, acc);                      // feat[:512] = h
  acc = gemm_k(attn_c + m * Hsz + half * 2, bp + (Hsz / 4) * 64, Hsz / 4, acc);// feat[512:] = attn_c
  int col = nt * 16 + l16;
#pragma unroll
  for (int v = 0; v < 8; v++) {
    int row = mt * 16 + half * 8 + v;
    out[(size_t)row * (Tsz * Csz) + t * Csz + col] = acc[v] + fc_b[col];
  }
}

// ---------------------------------------------------------------- deterministic l2_mean
__global__ void k_l2fin(const float* __restrict__ l2arr, float* __restrict__ dst) {
  __shared__ float red[256];
  int tid = threadIdx.x;
  float s = 0.0f;
  for (int i = tid; i < Tsz * Bsz; i += 256) s += l2arr[i];
  red[tid] = s; __syncthreads();
  for (int off = 128; off > 0; off >>= 1) { if (tid < off) red[tid] += red[tid + off]; __syncthreads(); }
  if (tid == 0) dst[0] = red[0] / ((float)Bsz * (float)(Tsz - 1));
}

// ---------------------------------------------------------------- host
extern "C" void kernel_launch(void* const* d_in, const int* in_sizes, int n_in,
                              void* d_out, int out_size, void* d_ws, size_t ws_size,
                              hipStream_t stream) {
  (void)in_sizes; (void)n_in; (void)out_size; (void)ws_size;
  const float* x      = (const float*)d_in[0];
  const float* w_ih   = (const float*)d_in[1];
  const float* w_hh   = (const float*)d_in[2];
  const float* b_ih   = (const float*)d_in[3];
  const float* b_hh   = (const float*)d_in[4];
  const float* w_t    = (const float*)d_in[5];
  const float* w_t2   = (const float*)d_in[6];
  const float* sel_w1 = (const float*)d_in[7];
  const float* fc_w   = (const float*)d_in[8];
  const float* fc_b   = (const float*)d_in[9];
  float* out = (float*)d_out;
  float* ws = (float*)d_ws;

  float* h       = ws;
  float* c       = h + Bsz * Hsz;
  float* h_old   = c + Bsz * Hsz;
  float* norms   = h_old + Bsz * RMAX * Hsz;
  float* g       = norms + Bsz * RMAX;
  float* predpre = g + Bsz * 4 * Hsz;
  float* hl      = predpre + Bsz * Hsz;
  float* uv      = hl + Bsz * Hsz;
  float* attn_c  = uv + (Bsz + Bsz * RMAX) * Hsz;
  float* a_g     = attn_c + Bsz * Hsz;
  float* araw    = a_g + Bsz * RMAX;
  float* l2arr   = araw + Bsz * RMAX;
  // packed weights (fragment order), built once per call, L2-resident afterwards
  float* pWg  = l2arr + Tsz * Bsz;          // 640  x 2048
  float* pWp  = pWg + 640 * 4 * Hsz;        // 512  x 512
  float* pWt1 = pWp + Hsz * Hsz;            // 512  x 512
  float* pWt2 = pWt1 + Hsz * Hsz;           // 512  x 512
  float* pWfc = pWt2 + Hsz * Hsz;           // 1024 x 128

  int initN = 2 * Bsz * Hsz + Bsz * RMAX * Hsz + Bsz * RMAX;  // h, c, h_old, norms
  k_init<<<(initN + 255) / 256, 256, 0, stream>>>(ws, initN);

  k_pack<0><<<(640 * 4 * Hsz + 255) / 256, 256, 0, stream>>>(w_ih, w_hh, pWg, 640, 4 * Hsz);
  k_pack<1><<<(Hsz * Hsz + 255) / 256, 256, 0, stream>>>(sel_w1, nullptr, pWp, Hsz, Hsz);
  k_pack<1><<<(Hsz * Hsz + 255) / 256, 256, 0, stream>>>(w_t, nullptr, pWt1, Hsz, Hsz);
  k_pack<1><<<(Hsz * Hsz + 255) / 256, 256, 0, stream>>>(w_t + Hsz * Hsz, nullptr, pWt2, Hsz, Hsz);
  k_pack<2><<<(2 * Hsz * Csz + 255) / 256, 256, 0, stream>>>(fc_w, nullptr, pWfc, 2 * Hsz, Csz);

  for (int t = 0; t < Tsz; t++) {
    int R = (t == 0) ? 1 : ((t < RMAX) ? t : RMAX);  // stored-state count at attend time
    k_gates<<<80, 256, 0, stream>>>(x, h, pWg, pWp, b_ih, b_hh, g, predpre, t);
    k_lstm<<<(Bsz * Hsz + 255) / 256, 256, 0, stream>>>(g, c, hl);
    int mtiles = 4 + 4 * R;
    k_uv<<<mtiles * 4, 256, 0, stream>>>(hl, h_old, pWt1, pWt2, uv, R);  // mtiles*32 waves
    k_attnraw<<<8 * R, 256, 0, stream>>>(uv, w_t2, araw, R);             // 64*R waves
    k_attend<<<Bsz, 256, 0, stream>>>(araw, h_old, hl, h, attn_c, a_g, R);
    if (t >= Tsz - 10)
      k_aviz<<<1, 32, 0, stream>>>(a_g, out + (size_t)Bsz * Tsz * Csz + (size_t)(t - (Tsz - 10)) * RMAX);
    k_select<<<Bsz, 256, 0, stream>>>(predpre, h, h_old, norms, l2arr, t);
    k_fc<<<4, 256, 0, stream>>>(h, attn_c, pWfc, fc_b, out, t);
  }
  k_l2fin<<<1, 256, 0, stream>>>(l2arr, out + (size_t)Bsz * Tsz * Csz + 10 * RMAX);
}